// DGCNN_64175401337639
// MI455X (gfx1250) — compile-verified
//
#include <hip/hip_runtime.h>

// CDNA5 / gfx1250 KNN (k=4) via V_WMMA_F32_16X16X4_F32.
// D = A(candidates 16x4) x B(queries 4x16): lane = query column,
// 8 acc VGPRs = candidate rows; lo/hi lane halves cover rows 0-7 / 8-15.

typedef float v2f __attribute__((ext_vector_type(2)));
typedef float v8f __attribute__((ext_vector_type(8)));

constexpr int BATCH = 4;
constexpr int NPTS  = 8192;
constexpr int KNN   = 4;

__global__ void xx_kernel(const float* __restrict__ x, float* __restrict__ xx, int total) {
    int i = blockIdx.x * blockDim.x + threadIdx.x;
    if (i < total) {
        float a = x[3 * i + 0], b = x[3 * i + 1], c = x[3 * i + 2];
        xx[i] = a * a + b * b + c * c;
    }
}

// Branch-free insert of (dv,di) into sorted-descending top-4 {v0..v3,i0..i3}.
// Tie-break: smaller index wins (matches lax.top_k ordering on equal values).
#define INS(dv, di)                                                         \
    {                                                                       \
        bool c = ((dv) > v3) || (((dv) == v3) && ((di) < i3));              \
        v3 = c ? (dv) : v3; i3 = c ? (di) : i3;                             \
        c = (v3 > v2) || ((v3 == v2) && (i3 < i2));                         \
        { float tv = v2; int ti = i2;                                       \
          v2 = c ? v3 : v2; i2 = c ? i3 : i2;                               \
          v3 = c ? tv : v3; i3 = c ? ti : i3; }                             \
        c = (v2 > v1) || ((v2 == v1) && (i2 < i1));                         \
        { float tv = v1; int ti = i1;                                       \
          v1 = c ? v2 : v1; i1 = c ? i2 : i1;                               \
          v2 = c ? tv : v2; i2 = c ? ti : i2; }                             \
        c = (v1 > v0) || ((v1 == v0) && (i1 < i0));                         \
        { float tv = v0; int ti = i0;                                       \
          v0 = c ? v1 : v0; i0 = c ? i1 : i0;                               \
          v1 = c ? tv : v1; i1 = c ? ti : i1; }                             \
    }

__global__ void __launch_bounds__(256)
knn_kernel(const float* __restrict__ x, const float* __restrict__ xx,
           float* __restrict__ out) {
    const int lane = threadIdx.x & 31;
    const int wave = (blockIdx.x * blockDim.x + threadIdx.x) >> 5;
    const int tilesPerBatch = NPTS / 16;               // 512
    const int b     = wave / tilesPerBatch;
    const int qbase = (wave % tilesPerBatch) * 16;

    const float* xb  = x  + (size_t)b * NPTS * 3;
    const float* xxb = xx + (size_t)b * NPTS;

    const bool hi = lane >= 16;
    const int  m  = lane & 15;

    // B operand (queries, 4x16 K-major): lanes 0-15 carry K=0,1 of query m;
    // lanes 16-31 carry K=2 and the zero pad (K=3). Loaded once, reused 512x.
    v2f bq;
    {
        const float* q = xb + 3 * (qbase + m);
        if (!hi) { bq.x = q[0]; bq.y = q[1]; }
        else     { bq.x = q[2]; bq.y = 0.0f; }
    }
    const float xxq = xxb[qbase + m];

    float v0 = -1e30f, v1 = -1e30f, v2 = -1e30f, v3 = -1e30f;
    int   i0 = 0x7fffffff, i1 = 0x7fffffff, i2 = 0x7fffffff, i3 = 0x7fffffff;

    const int off = hi ? 8 : 0;

    for (int cb = 0; cb < NPTS; cb += 16) {
        // A operand (candidates, 16x4): same per-lane K split as B.
        v2f a;
        const float* p = xb + 3 * (cb + m);
        if (!hi) { a.x = p[0]; a.y = p[1]; }
        else     { a.x = p[2]; a.y = 0.0f; }

        // ||candidate||^2 for the 8 rows this lane reduces (L2-resident).
        const float4* xc4 = (const float4*)(xxb + cb + off);
        float4 xcA = xc4[0];
        float4 xcB = xc4[1];

        v8f acc = {};
        acc = __builtin_amdgcn_wmma_f32_16x16x4_f32(
                  /*neg_a=*/false, a, /*neg_b=*/false, bq,
                  /*c_mod=*/(short)0, acc, /*reuse_a=*/false, /*reuse_b=*/false);

        float xxc[8] = {xcA.x, xcA.y, xcA.z, xcA.w, xcB.x, xcB.y, xcB.z, xcB.w};
#pragma unroll
        for (int r = 0; r < 8; ++r) {
            float d  = 2.0f * acc[r] - xxc[r] - xxq;   // -squared distance
            int   ci = cb + off + r;
            INS(d, ci);
        }
    }

    // Merge with partner lane (other half of candidate rows) — wave32 xor-16.
    float ov0 = __shfl_xor(v0, 16, 32), ov1 = __shfl_xor(v1, 16, 32);
    float ov2 = __shfl_xor(v2, 16, 32), ov3 = __shfl_xor(v3, 16, 32);
    int   oi0 = __shfl_xor(i0, 16, 32), oi1 = __shfl_xor(i1, 16, 32);
    int   oi2 = __shfl_xor(i2, 16, 32), oi3 = __shfl_xor(i3, 16, 32);
    INS(ov0, oi0); INS(ov1, oi1); INS(ov2, oi2); INS(ov3, oi3);

    if (!hi) {
        const int q = qbase + m;
        float* o = out + ((size_t)b * NPTS + q) * KNN * 3;
        int idxs[4] = {i0, i1, i2, i3};
#pragma unroll
        for (int j = 0; j < KNN; ++j) {
            const float* src = xb + 3 * idxs[j];
            o[3 * j + 0] = src[0];
            o[3 * j + 1] = src[1];
            o[3 * j + 2] = src[2];
        }
    }
}

extern "C" void kernel_launch(void* const* d_in, const int* in_sizes, int n_in,
                              void* d_out, int out_size, void* d_ws, size_t ws_size,
                              hipStream_t stream) {
    const float* x  = (const float*)d_in[0];
    float* xx  = (float*)d_ws;              // BATCH*NPTS floats = 128 KB
    float* out = (float*)d_out;

    const int total = BATCH * NPTS;         // 32768 points
    xx_kernel<<<(total + 255) / 256, 256, 0, stream>>>(x, xx, total);

    const int waves   = BATCH * (NPTS / 16);            // 2048 query tiles
    const int threads = 256;                            // 8 waves per block
    const int blocks  = (waves * 32) / threads;         // 256 blocks
    knn_kernel<<<blocks, threads, 0, stream>>>(x, xx, out);
}